// GAT2_82575041232960
// MI455X (gfx1250) — compile-verified
//
#include <hip/hip_runtime.h>
#include <math.h>

// ---------------------------------------------------------------------------
// GATConv (concat=False, 4 heads, 128 -> 64) + BatchNorm + ELU for gfx1250.
// Phases:
//   0) init: m=-inf, denom/out_hc/sums = 0
//   1) x_proj = x @ W via V_WMMA_F32_16X16X4_F32 (fp32 WMMA)
//   2) a_src/a_dst = per-(node,head) dot(x_proj, att)
//   3) edge pass A: segment-max of leaky_relu(a_src[s]+a_dst[d]) per dst
//   4) edge pass B: ex = exp(e - m[dst]); segment-sum -> denom
//   5) edge pass C: out_hc[dst] += (ex/denom[dst]) * x_proj[src]  (wave/edge)
//   6) head mean + bias, per-channel sum/sumsq (BN batch stats)
//   7) BN normalize + ELU -> d_out
// ---------------------------------------------------------------------------

typedef float v2f __attribute__((ext_vector_type(2)));
typedef float v8f __attribute__((ext_vector_type(8)));

#define KDIM 128          // IN_CH
#define NOUT 256          // HEADS * OUT_CH
#define OUTC 64
#define LDA  (KDIM + 4)   // LDS row pitch (floats), pad to avoid bank conflicts
#define NEG_SLOPE 0.2f
#define BN_EPS 1e-5f

// ---------------------------------------------------------------------------
// 0) workspace init
// ---------------------------------------------------------------------------
__global__ void gat_init_kernel(float* __restrict__ mneg, size_t m_count,
                                float* __restrict__ zptr, size_t z_count) {
    size_t i = (size_t)blockIdx.x * blockDim.x + threadIdx.x;
    size_t stride = (size_t)gridDim.x * blockDim.x;
    for (size_t j = i; j < m_count; j += stride) mneg[j] = -INFINITY;
    for (size_t j = i; j < z_count; j += stride) zptr[j] = 0.0f;
}

// ---------------------------------------------------------------------------
// 1) x_proj = x @ W  (M=n, K=128, N=256), fp32 WMMA 16x16x4
//    block = 256 threads = 8 waves; block owns a 16-row M tile, each wave
//    owns two 16-wide N tiles (covers all 256 cols).
// ---------------------------------------------------------------------------
__global__ __launch_bounds__(256)
void gat_gemm_proj_kernel(const float* __restrict__ x, const float* __restrict__ W,
                          float* __restrict__ xp, int n) {
    __shared__ float As[16 * LDA];
    const int m0  = blockIdx.x * 16;
    const int tid = threadIdx.x;

    // Cooperative load of A tile: 16 x 128 floats (8 per thread, two float4)
    {
        const int row = tid >> 4;         // 0..15
        const int col = (tid & 15) * 8;   // 0..120
        const int gr  = m0 + row;
        float4 v0 = make_float4(0.f, 0.f, 0.f, 0.f), v1 = v0;
        if (gr < n) {
            const float4* p = (const float4*)(x + (size_t)gr * KDIM + col);
            v0 = p[0]; v1 = p[1];
        }
        float* dst = As + row * LDA + col;
        *(float4*)(dst)     = v0;
        *(float4*)(dst + 4) = v1;
    }
    __syncthreads();

    const int wave = tid >> 5;     // 0..7
    const int lane = tid & 31;
    const int half = lane >> 4;    // 0: K=0,1   1: K=2,3
    const int l16  = lane & 15;

    const int n0 = wave * 32;
    const float* Wc0 = W + (n0 + l16);        // cols n0..n0+15
    const float* Wc1 = W + (n0 + 16 + l16);   // cols n0+16..n0+31
    const float* Ar  = As + l16 * LDA;        // A row M = l16

    v8f acc0 = {};
    v8f acc1 = {};

    for (int k = 0; k < KDIM; k += 4) {
        const int kk = k + half * 2;
        v2f a;                                     // A 16x4 fragment
        a.x = Ar[kk];
        a.y = Ar[kk + 1];
        v2f b0, b1;                                // B 4x16 fragments
        b0.x = Wc0[(size_t)kk * NOUT];
        b0.y = Wc0[(size_t)(kk + 1) * NOUT];
        b1.x = Wc1[(size_t)kk * NOUT];
        b1.y = Wc1[(size_t)(kk + 1) * NOUT];
        acc0 = __builtin_amdgcn_wmma_f32_16x16x4_f32(false, a, false, b0,
                                                     (short)0, acc0, false, false);
        acc1 = __builtin_amdgcn_wmma_f32_16x16x4_f32(false, a, false, b1,
                                                     (short)0, acc1, false, false);
    }

    // D layout: VGPR r -> M=r (lanes 0-15) / M=r+8 (lanes 16-31); N = l16
    #pragma unroll
    for (int r = 0; r < 8; ++r) {
        const int gm = m0 + r + half * 8;
        if (gm < n) {
            xp[(size_t)gm * NOUT + n0 + l16]      = acc0[r];
            xp[(size_t)gm * NOUT + n0 + 16 + l16] = acc1[r];
        }
    }
}

// ---------------------------------------------------------------------------
// 2) attention coefficients: one wave per (node, head)
// ---------------------------------------------------------------------------
__global__ __launch_bounds__(256)
void gat_acoef_kernel(const float* __restrict__ xp, const float* __restrict__ att_src,
                      const float* __restrict__ att_dst, float* __restrict__ a_src,
                      float* __restrict__ a_dst, int n) {
    const int tid  = threadIdx.x;
    const int wave = tid >> 5;                 // 0..7
    const int lane = tid & 31;
    const int node = blockIdx.x * 2 + (wave >> 2);
    const int h    = wave & 3;
    if (node >= n) return;

    const float* row = xp + (size_t)node * NOUT + h * OUTC;
    const float* as  = att_src + h * OUTC;
    const float* ad  = att_dst + h * OUTC;
    float x0 = row[lane], x1 = row[lane + 32];
    float s = x0 * as[lane] + x1 * as[lane + 32];
    float d = x0 * ad[lane] + x1 * ad[lane + 32];
    #pragma unroll
    for (int off = 16; off > 0; off >>= 1) {
        s += __shfl_down(s, off, 32);
        d += __shfl_down(d, off, 32);
    }
    if (lane == 0) {
        a_src[(size_t)node * 4 + h] = s;
        a_dst[(size_t)node * 4 + h] = d;
    }
}

// ---------------------------------------------------------------------------
// 3) edge pass A: segment-max (float atomic max -> global_atomic_max_num_f32)
// ---------------------------------------------------------------------------
__global__ __launch_bounds__(256)
void gat_edge_max_kernel(const long long* __restrict__ ei, long long E,
                         const float* __restrict__ a_src, const float* __restrict__ a_dst,
                         float* __restrict__ mmax, long long n) {
    long long e = (size_t)blockIdx.x * blockDim.x + threadIdx.x;
    if (e >= E + n) return;
    long long s, d;
    if (e < E) { s = ei[e]; d = ei[E + e]; } else { s = e - E; d = s; }
    float4 vs = *(const float4*)(a_src + s * 4);
    float4 vd = *(const float4*)(a_dst + d * 4);
    float ev[4] = {vs.x + vd.x, vs.y + vd.y, vs.z + vd.z, vs.w + vd.w};
    #pragma unroll
    for (int h = 0; h < 4; ++h) {
        float v = ev[h];
        v = v > 0.f ? v : NEG_SLOPE * v;
        atomicMax(&mmax[d * 4 + h], v);
    }
}

// ---------------------------------------------------------------------------
// 4) edge pass B: ex = exp(e - m[dst]); denom[dst] += ex
// ---------------------------------------------------------------------------
__global__ __launch_bounds__(256)
void gat_edge_exp_kernel(const long long* __restrict__ ei, long long E,
                         const float* __restrict__ a_src, const float* __restrict__ a_dst,
                         const float* __restrict__ mmax, float* __restrict__ exbuf,
                         float* __restrict__ denom, long long n) {
    long long e = (size_t)blockIdx.x * blockDim.x + threadIdx.x;
    if (e >= E + n) return;
    long long s, d;
    if (e < E) { s = ei[e]; d = ei[E + e]; } else { s = e - E; d = s; }
    float4 vs = *(const float4*)(a_src + s * 4);
    float4 vd = *(const float4*)(a_dst + d * 4);
    float4 vm = *(const float4*)(mmax + d * 4);
    float ev[4] = {vs.x + vd.x, vs.y + vd.y, vs.z + vd.z, vs.w + vd.w};
    float mv[4] = {vm.x, vm.y, vm.z, vm.w};
    float4 exo;
    float* exp_out = (float*)&exo;
    #pragma unroll
    for (int h = 0; h < 4; ++h) {
        float v = ev[h];
        v = v > 0.f ? v : NEG_SLOPE * v;
        float ex = __expf(v - mv[h]);
        exp_out[h] = ex;
        atomicAdd(&denom[d * 4 + h], ex);
    }
    *(float4*)(exbuf + e * 4) = exo;
}

// ---------------------------------------------------------------------------
// 5) edge pass C: out_hc[dst] += alpha * x_proj[src]; one wave per edge,
//    each lane owns 8 contiguous channels (exactly one head per lane).
// ---------------------------------------------------------------------------
__global__ __launch_bounds__(256)
void gat_edge_msg_kernel(const long long* __restrict__ ei, long long E,
                         const float* __restrict__ exbuf, const float* __restrict__ denom,
                         const float* __restrict__ xp, float* __restrict__ out_hc,
                         long long n) {
    const int lane = threadIdx.x & 31;
    long long e = (size_t)blockIdx.x * (blockDim.x >> 5) + (threadIdx.x >> 5);
    if (e >= E + n) return;
    long long s, d;
    if (e < E) { s = ei[e]; d = ei[E + e]; } else { s = e - E; d = s; }

    float4 exv = *(const float4*)(exbuf + e * 4);
    float4 dnv = *(const float4*)(denom + d * 4);
    float alpha[4] = {exv.x / dnv.x, exv.y / dnv.y, exv.z / dnv.z, exv.w / dnv.w};

    const int   ch   = lane * 8;          // 8 contiguous channels, head = lane>>3
    const float a    = alpha[lane >> 3];
    const float* xr  = xp + (size_t)s * NOUT + ch;
    float*       orw = out_hc + (size_t)d * NOUT + ch;
    float4 v0 = *(const float4*)(xr);
    float4 v1 = *(const float4*)(xr + 4);
    atomicAdd(&orw[0], a * v0.x);
    atomicAdd(&orw[1], a * v0.y);
    atomicAdd(&orw[2], a * v0.z);
    atomicAdd(&orw[3], a * v0.w);
    atomicAdd(&orw[4], a * v1.x);
    atomicAdd(&orw[5], a * v1.y);
    atomicAdd(&orw[6], a * v1.z);
    atomicAdd(&orw[7], a * v1.w);
}

// ---------------------------------------------------------------------------
// 6) head mean + bias, per-channel BN batch stats (sum / sumsq)
// ---------------------------------------------------------------------------
__global__ __launch_bounds__(64)
void gat_stats_kernel(const float* __restrict__ out_hc, const float* __restrict__ bias,
                      float* __restrict__ out_mean, float* __restrict__ sums, int n) {
    const int c = threadIdx.x;            // 0..63
    const float b = bias[c];
    float ls = 0.f, lq = 0.f;
    for (int node = blockIdx.x; node < n; node += gridDim.x) {
        const float* row = out_hc + (size_t)node * NOUT;
        float v = 0.25f * (row[c] + row[c + 64] + row[c + 128] + row[c + 192]) + b;
        out_mean[(size_t)node * OUTC + c] = v;
        ls += v;
        lq += v * v;
    }
    atomicAdd(&sums[c], ls);
    atomicAdd(&sums[64 + c], lq);
}

// ---------------------------------------------------------------------------
// 7) BatchNorm (batch stats, biased var) + ELU
// ---------------------------------------------------------------------------
__global__ __launch_bounds__(256)
void gat_bn_elu_kernel(const float* __restrict__ out_mean, const float* __restrict__ sums,
                       const float* __restrict__ bn_w, const float* __restrict__ bn_b,
                       float* __restrict__ out, int n) {
    long long i = (size_t)blockIdx.x * blockDim.x + threadIdx.x;
    if (i >= (long long)n * OUTC) return;
    const int c = (int)(i & 63);
    const float inv_n = 1.0f / (float)n;
    const float mu   = sums[c] * inv_n;
    const float var  = sums[64 + c] * inv_n - mu * mu;
    const float rstd = rsqrtf(var + BN_EPS);
    float v = (out_mean[i] - mu) * rstd * bn_w[c] + bn_b[c];
    out[i] = v > 0.f ? v : expm1f(v);
}

// ---------------------------------------------------------------------------
// launcher
// ---------------------------------------------------------------------------
extern "C" void kernel_launch(void* const* d_in, const int* in_sizes, int n_in,
                              void* d_out, int out_size, void* d_ws, size_t ws_size,
                              hipStream_t stream) {
    const float*     x       = (const float*)d_in[0];
    const long long* ei      = (const long long*)d_in[1];   // int64 edge_index [2,E]
    const float*     W       = (const float*)d_in[2];
    const float*     att_src = (const float*)d_in[3];
    const float*     att_dst = (const float*)d_in[4];
    const float*     bias    = (const float*)d_in[5];
    const float*     bn_w    = (const float*)d_in[6];
    const float*     bn_b    = (const float*)d_in[7];
    float*           out     = (float*)d_out;

    const long long n   = in_sizes[0] / KDIM;   // 50000
    const long long E   = in_sizes[1] / 2;      // 800000
    const long long tot = E + n;                // self loops appended

    // workspace layout (floats); denom..sums contiguous so one zero region
    float* ws = (float*)d_ws;
    size_t off = 0;
    float* xp     = ws + off; off += (size_t)n * NOUT;      // [N,256]
    float* exbuf  = ws + off; off += (size_t)tot * 4;       // [E+N,4]
    float* a_src  = ws + off; off += (size_t)n * 4;         // [N,4]
    float* a_dst  = ws + off; off += (size_t)n * 4;         // [N,4]
    float* mmax   = ws + off; off += (size_t)n * 4;         // [N,4] -> -inf
    float* zbase  = ws + off;                               // zero region start
    float* denom  = ws + off; off += (size_t)n * 4;         // [N,4]
    float* out_hc = ws + off; off += (size_t)n * NOUT;      // [N,256]
    float* sums   = ws + off; off += 128;                   // sum[64], sumsq[64]
    const size_t zcount = (size_t)n * 4 + (size_t)n * NOUT + 128;
    float* out_mean = ws + off; off += (size_t)n * OUTC;    // [N,64]
    (void)ws_size; (void)n_in; (void)out_size;

    gat_init_kernel<<<2048, 256, 0, stream>>>(mmax, (size_t)n * 4, zbase, zcount);

    gat_gemm_proj_kernel<<<(int)((n + 15) / 16), 256, 0, stream>>>(x, W, xp, (int)n);

    gat_acoef_kernel<<<(int)((n + 1) / 2), 256, 0, stream>>>(xp, att_src, att_dst,
                                                             a_src, a_dst, (int)n);

    gat_edge_max_kernel<<<(int)((tot + 255) / 256), 256, 0, stream>>>(
        ei, E, a_src, a_dst, mmax, n);

    gat_edge_exp_kernel<<<(int)((tot + 255) / 256), 256, 0, stream>>>(
        ei, E, a_src, a_dst, mmax, exbuf, denom, n);

    gat_edge_msg_kernel<<<(int)((tot + 7) / 8), 256, 0, stream>>>(
        ei, E, exbuf, denom, xp, out_hc, n);

    gat_stats_kernel<<<1024, 64, 0, stream>>>(out_hc, bias, out_mean, sums, (int)n);

    gat_bn_elu_kernel<<<(int)(((long long)n * OUTC + 255) / 256), 256, 0, stream>>>(
        out_mean, sums, bn_w, bn_b, out, (int)n);
}